// SRCF_63471026700702
// MI455X (gfx1250) — compile-verified
//
#include <hip/hip_runtime.h>
#include <math.h>

typedef float v2f __attribute__((ext_vector_type(2)));
typedef float v8f __attribute__((ext_vector_type(8)));

#define N_USER 60000
#define N_ITEM 120000
#define NTOT   (N_USER + N_ITEM)
#define EMB    64
#define BATCH  4096
#define NNZ_ADJ 3600000
#define NNZ_UG  960000
#define NNZ_IG  1920000

// ---------------------------------------------------------------------------
// Zero a float buffer (grid-stride)
// ---------------------------------------------------------------------------
__global__ void zero_f32(float* __restrict__ p, long n) {
  long i = (long)blockIdx.x * blockDim.x + threadIdx.x;
  long stride = (long)gridDim.x * blockDim.x;
  for (; i < n; i += stride) p[i] = 0.0f;
}

// ---------------------------------------------------------------------------
// all_emb[:, 0:64] = concat(user_emb, item_emb)   (row stride 256)
// ---------------------------------------------------------------------------
__global__ void copy_embed(const float* __restrict__ ue,
                           const float* __restrict__ ie,
                           float* __restrict__ all_emb) {
  long i = (long)blockIdx.x * blockDim.x + threadIdx.x;
  if (i >= (long)NTOT * 64) return;
  int row = (int)(i >> 6);
  int j = (int)(i & 63);
  float v = (row < N_USER) ? ue[(long)row * 64 + j]
                           : ie[(long)(row - N_USER) * 64 + j];
  all_emb[(long)row * 256 + j] = v;
}

// ---------------------------------------------------------------------------
// SpMM scatter: y[rows[e], :] += vals[e] * x[cols[e], :]
// one 32-lane wave per edge, each lane does cols {lane, lane+32}
// ---------------------------------------------------------------------------
__global__ void spmm_scatter(const int* __restrict__ rows,
                             const int* __restrict__ cols,
                             const float* __restrict__ vals,
                             const float* __restrict__ x, int ldx,
                             float* __restrict__ y, int ldy, int nnz) {
  int lane = threadIdx.x & 31;
  int e = blockIdx.x * (blockDim.x >> 5) + (threadIdx.x >> 5);
  if (e >= nnz) return;
  int r = rows[e];
  int c = cols[e];
  float v = vals[e];
  const float* xs = x + (long)c * ldx;
  float* yd = y + (long)r * ldy;
  atomicAdd(yd + lane,      v * xs[lane]);
  atomicAdd(yd + lane + 32, v * xs[lane + 32]);
}

// ---------------------------------------------------------------------------
// WMMA f32 GEMM over 64-wide matrices:
//   DUAL=0:  D = act( A @ W1 + bias1 )
//   DUAL=1:  D = act( A @ W1 + (A .* E) @ W2 + bias1 + bias2 )
// ACT: 0 = identity, 1 = ELU, 2 = leaky_relu(0.2)
// One wave computes one 16x16 output tile; K=64 -> 16 (or 32) V_WMMA_F32_16X16X4_F32.
// A-frag layout (ISA 7.12.2, 32-bit A 16x4): lanes 0-15 hold K=0,1; lanes 16-31 K=2,3.
// B-frag (4x16): VGPR0 lanes0-15=K0 / lanes16-31=K2; VGPR1 = K1/K3.
// C/D (16x16 f32): VGPR r -> row r (lanes0-15) / row r+8 (lanes16-31), col = lane%16.
// ---------------------------------------------------------------------------
template <int ACT, bool DUAL>
__global__ void gemm64_wmma(const float* __restrict__ A, int lda,
                            const float* __restrict__ E, int lde,
                            const float* __restrict__ W1,
                            const float* __restrict__ bias1,
                            const float* __restrict__ W2,
                            const float* __restrict__ bias2,
                            float* __restrict__ D, int ldd, int nrows) {
  int wave = threadIdx.x >> 5;              // 0..7
  int lane = threadIdx.x & 31;
  int col_tile = wave & 3;                  // 4 column tiles cover 64 outputs
  int row_tile = blockIdx.x * 2 + (wave >> 2);
  int m0 = row_tile * 16;
  if (m0 >= nrows) return;
  int n0 = col_tile * 16;
  int half = lane >> 4;                     // 0: K pair {0,1}; 1: K pair {2,3}
  int lm = lane & 15;
  int m = m0 + lm;                          // A-frag row for this lane
  int n = n0 + lm;                          // B/C/D column for this lane

  const float* arow = A + (long)m * lda;
  const float* erow = DUAL ? (E + (long)m * lde) : nullptr;

  v8f acc = {};
#pragma unroll
  for (int kk = 0; kk < 16; ++kk) {
    int k2 = kk * 4 + half * 2;
    v2f a;
    a.x = arow[k2];
    a.y = arow[k2 + 1];
    v2f b1;
    b1.x = W1[k2 * 64 + n];
    b1.y = W1[(k2 + 1) * 64 + n];
    acc = __builtin_amdgcn_wmma_f32_16x16x4_f32(false, a, false, b1,
                                                (short)0, acc, false, false);
    if (DUAL) {
      v2f a2;
      a2.x = a.x * erow[k2];
      a2.y = a.y * erow[k2 + 1];
      v2f b2;
      b2.x = W2[k2 * 64 + n];
      b2.y = W2[(k2 + 1) * 64 + n];
      acc = __builtin_amdgcn_wmma_f32_16x16x4_f32(false, a2, false, b2,
                                                  (short)0, acc, false, false);
    }
  }

  float bsum = bias1[n] + (DUAL ? bias2[n] : 0.0f);
#pragma unroll
  for (int r = 0; r < 8; ++r) {
    int mr = m0 + half * 8 + r;
    float v = acc[r] + bsum;
    if (ACT == 1) v = (v > 0.0f) ? v : (expf(v) - 1.0f);        // ELU
    else if (ACT == 2) v = (v > 0.0f) ? v : 0.2f * v;           // leaky
    D[(long)mr * ldd + n] = v;
  }
}

// ---------------------------------------------------------------------------
// In-place row-wise L2 normalize (64 elements/row, arbitrary row stride)
// one wave per row
// ---------------------------------------------------------------------------
__global__ void l2norm_rows(float* __restrict__ p, int ld, int nrows) {
  int lane = threadIdx.x & 31;
  int row = blockIdx.x * (blockDim.x >> 5) + (threadIdx.x >> 5);
  if (row >= nrows) return;
  float* r = p + (long)row * ld;
  float a = r[lane];
  float b = r[lane + 32];
  float s = a * a + b * b;
#pragma unroll
  for (int off = 16; off; off >>= 1) s += __shfl_xor(s, off, 32);
  float inv = 1.0f / fmaxf(sqrtf(s), 1e-12f);
  r[lane] = a * inv;
  r[lane + 32] = b * inv;
}

// ---------------------------------------------------------------------------
// Final gather: out = concat( user_full[users], item_full[pos], item_full[neg] )
// user_full = [all_emb_row(256) | relu(u_h row 64)]  (320 cols)
// ---------------------------------------------------------------------------
__global__ void gather_out(const int* __restrict__ users,
                           const int* __restrict__ pos,
                           const int* __restrict__ neg,
                           const float* __restrict__ all_emb,
                           const float* __restrict__ u_h,
                           const float* __restrict__ i_h,
                           float* __restrict__ out) {
  long total = 3L * BATCH * 320;
  long i = (long)blockIdx.x * blockDim.x + threadIdx.x;
  if (i >= total) return;
  int group = (int)(i / ((long)BATCH * 320));
  int rem = (int)(i - (long)group * BATCH * 320);
  int b = rem / 320;
  int j = rem - b * 320;
  int idx = (group == 0) ? users[b] : (group == 1 ? pos[b] : neg[b]);
  int grow = (group == 0) ? idx : (N_USER + idx);
  float v;
  if (j < 256) {
    v = all_emb[(long)grow * 256 + j];
  } else {
    const float* h = (group == 0) ? u_h : i_h;
    float hv = h[(long)idx * 64 + (j - 256)];
    v = hv > 0.0f ? hv : 0.0f;
  }
  out[i] = v;
}

// ---------------------------------------------------------------------------
extern "C" void kernel_launch(void* const* d_in, const int* in_sizes, int n_in,
                              void* d_out, int out_size, void* d_ws, size_t ws_size,
                              hipStream_t stream) {
  const int* users = (const int*)d_in[0];
  const int* pos_items = (const int*)d_in[1];
  const int* neg_items = (const int*)d_in[2];
  const int* adj_rows = (const int*)d_in[3];
  const int* adj_cols = (const int*)d_in[4];
  const float* adj_vals = (const float*)d_in[5];
  const int* ug_rows = (const int*)d_in[6];
  const int* ug_cols = (const int*)d_in[7];
  const float* ug_vals = (const float*)d_in[8];
  const int* ig_rows = (const int*)d_in[9];
  const int* ig_cols = (const int*)d_in[10];
  const float* ig_vals = (const float*)d_in[11];
  const float* user_emb = (const float*)d_in[12];
  const float* item_emb = (const float*)d_in[13];
  const float* W_gc = (const float*)d_in[14];  // (3,64,64)
  const float* b_gc = (const float*)d_in[15];  // (3,1,64)
  const float* W_bi = (const float*)d_in[16];
  const float* b_bi = (const float*)d_in[17];
  const float* Wu0 = (const float*)d_in[18];
  const float* bu0 = (const float*)d_in[19];
  const float* Wu1 = (const float*)d_in[20];
  const float* bu1 = (const float*)d_in[21];
  const float* Wi0 = (const float*)d_in[22];
  const float* bi0 = (const float*)d_in[23];
  const float* Wi1 = (const float*)d_in[24];
  const float* bi1 = (const float*)d_in[25];

  // workspace layout (floats)
  float* ws = (float*)d_ws;
  float* all_emb = ws;                                   // NTOT x 256
  float* side    = all_emb + (long)NTOT * 256;           // NTOT x 64
  float* tmp     = side + (long)NTOT * 64;               // N_ITEM x 64 (MLP scratch)
  float* u_h     = tmp + (long)N_ITEM * 64;              // N_USER x 64
  float* i_h     = u_h + (long)N_USER * 64;              // N_ITEM x 64

  const int TB = 256;
  const int spmm_adj_blocks = (NNZ_ADJ + 7) / 8;   // 8 edges / 256-thread block
  const int spmm_ug_blocks  = (NNZ_UG + 7) / 8;
  const int spmm_ig_blocks  = (NNZ_IG + 7) / 8;
  const int gemm_all_blocks  = (NTOT / 16 + 1) / 2;    // 2 row tiles per block
  const int gemm_user_blocks = (N_USER / 16 + 1) / 2;
  const int gemm_item_blocks = (N_ITEM / 16 + 1) / 2;

  // ego0 -> all_emb slot 0 (raw, never normalized, matching reference)
  copy_embed<<<((long)NTOT * 64 + TB - 1) / TB, TB, 0, stream>>>(user_emb, item_emb, all_emb);

  // ---- NGCF layer 0: side = A @ slot0 ; slot1 = leaky(side@Wgc0 + (slot0*side)@Wbi0 + b)
  zero_f32<<<4096, TB, 0, stream>>>(side, (long)NTOT * 64);
  spmm_scatter<<<spmm_adj_blocks, TB, 0, stream>>>(adj_rows, adj_cols, adj_vals,
                                                   all_emb + 0, 256, side, 64, NNZ_ADJ);
  gemm64_wmma<2, true><<<gemm_all_blocks, TB, 0, stream>>>(
      side, 64, all_emb + 0, 256,
      W_gc + 0 * 64 * 64, b_gc + 0 * 64, W_bi + 0 * 64 * 64, b_bi + 0 * 64,
      all_emb + 64, 256, NTOT);

  // ---- layer 1 (reads unnormalized slot1, then slot1 is normalized in place)
  zero_f32<<<4096, TB, 0, stream>>>(side, (long)NTOT * 64);
  spmm_scatter<<<spmm_adj_blocks, TB, 0, stream>>>(adj_rows, adj_cols, adj_vals,
                                                   all_emb + 64, 256, side, 64, NNZ_ADJ);
  gemm64_wmma<2, true><<<gemm_all_blocks, TB, 0, stream>>>(
      side, 64, all_emb + 64, 256,
      W_gc + 1 * 64 * 64, b_gc + 1 * 64, W_bi + 1 * 64 * 64, b_bi + 1 * 64,
      all_emb + 128, 256, NTOT);
  l2norm_rows<<<(NTOT + 7) / 8, TB, 0, stream>>>(all_emb + 64, 256, NTOT);

  // ---- layer 2
  zero_f32<<<4096, TB, 0, stream>>>(side, (long)NTOT * 64);
  spmm_scatter<<<spmm_adj_blocks, TB, 0, stream>>>(adj_rows, adj_cols, adj_vals,
                                                   all_emb + 128, 256, side, 64, NNZ_ADJ);
  gemm64_wmma<2, true><<<gemm_all_blocks, TB, 0, stream>>>(
      side, 64, all_emb + 128, 256,
      W_gc + 2 * 64 * 64, b_gc + 2 * 64, W_bi + 2 * 64 * 64, b_bi + 2 * 64,
      all_emb + 192, 256, NTOT);
  l2norm_rows<<<(NTOT + 7) / 8, TB, 0, stream>>>(all_emb + 128, 256, NTOT);
  l2norm_rows<<<(NTOT + 7) / 8, TB, 0, stream>>>(all_emb + 192, 256, NTOT);

  // ---- user MLP path: u_h = spmm(UG, elu(spmm(UG, ue)@Wu0+bu0)) @ Wu1 + bu1
  zero_f32<<<2048, TB, 0, stream>>>(tmp, (long)N_USER * 64);
  spmm_scatter<<<spmm_ug_blocks, TB, 0, stream>>>(ug_rows, ug_cols, ug_vals,
                                                  user_emb, 64, tmp, 64, NNZ_UG);
  gemm64_wmma<1, false><<<gemm_user_blocks, TB, 0, stream>>>(
      tmp, 64, nullptr, 0, Wu0, bu0, nullptr, nullptr, side, 64, N_USER);
  zero_f32<<<2048, TB, 0, stream>>>(tmp, (long)N_USER * 64);
  spmm_scatter<<<spmm_ug_blocks, TB, 0, stream>>>(ug_rows, ug_cols, ug_vals,
                                                  side, 64, tmp, 64, NNZ_UG);
  gemm64_wmma<0, false><<<gemm_user_blocks, TB, 0, stream>>>(
      tmp, 64, nullptr, 0, Wu1, bu1, nullptr, nullptr, u_h, 64, N_USER);

  // ---- item MLP path
  zero_f32<<<2048, TB, 0, stream>>>(tmp, (long)N_ITEM * 64);
  spmm_scatter<<<spmm_ig_blocks, TB, 0, stream>>>(ig_rows, ig_cols, ig_vals,
                                                  item_emb, 64, tmp, 64, NNZ_IG);
  gemm64_wmma<1, false><<<gemm_item_blocks, TB, 0, stream>>>(
      tmp, 64, nullptr, 0, Wi0, bi0, nullptr, nullptr, side, 64, N_ITEM);
  zero_f32<<<2048, TB, 0, stream>>>(tmp, (long)N_ITEM * 64);
  spmm_scatter<<<spmm_ig_blocks, TB, 0, stream>>>(ig_rows, ig_cols, ig_vals,
                                                  side, 64, tmp, 64, NNZ_IG);
  gemm64_wmma<0, false><<<gemm_item_blocks, TB, 0, stream>>>(
      tmp, 64, nullptr, 0, Wi1, bi1, nullptr, nullptr, i_h, 64, N_ITEM);

  // ---- final gather (3 x 4096 x 320)
  long total = 3L * BATCH * 320;
  gather_out<<<(total + TB - 1) / TB, TB, 0, stream>>>(
      users, pos_items, neg_items, all_emb, u_h, i_h, (float*)d_out);
}